// MultiModalFusionGAT_11759620456962
// MI455X (gfx1250) — compile-verified
//
#include <hip/hip_runtime.h>
#include <math.h>

#define C128 128
#define HEADS 2
#define DHEAD 64

typedef float v2f __attribute__((ext_vector_type(2)));
typedef float v8f __attribute__((ext_vector_type(8)));

// ---------------------------------------------------------------------------
// fp32 WMMA GEMM core, LDS-staged B.
//
// W staged into LDS in k-pair layout:
//   Wp[kp][n] = { W[2kp][n], W[2kp+1][n] }   (8 bytes, one ds_load_b64)
// with odd-kp rows rotated by 16 columns so lanes 0-15 (even kp) and lanes
// 16-31 (odd kp) hit disjoint LDS bank groups -> conflict-free b64 reads.
//
// wmma_tiles<NT>: one wave computes NT 16-row x 128-col tiles, branch-free.
// NT=2 for full waves (8 B-frags feed 16 WMMAs), NT=1 for the tail tile.
// ---------------------------------------------------------------------------
#define WL_FLOATS 16384  // 128x128 f32 = 64KB

__device__ __forceinline__ void fill_W_lds(const float* __restrict__ W,
                                           float* __restrict__ Wl, int tid, int nthr) {
  for (int g4 = tid; g4 < 4096; g4 += nthr) {
    const int k  = g4 >> 5;          // row 0..127
    const int n0 = (g4 & 31) * 4;    // col group
    const float4 w = *(const float4*)(W + (size_t)k * C128 + n0);
    const int kp = k >> 1, j = k & 1;
    const int rot = (kp & 1) ? 16 : 0;
    float vals[4] = {w.x, w.y, w.z, w.w};
#pragma unroll
    for (int u = 0; u < 4; ++u) {
      const int nn = (n0 + u + rot) & 127;
      Wl[kp * 256 + nn * 2 + j] = vals[u];
    }
  }
}

template <int NT>
__device__ __forceinline__ void wmma_tiles(const float* __restrict__ X,
                                           const float* __restrict__ Wl,
                                           int row0, int lane, v8f (&acc)[NT][8]) {
  const int half = lane >> 4;
  const int l16  = lane & 15;
  const float* arow[NT];
#pragma unroll
  for (int t = 0; t < NT; ++t)
    arow[t] = X + (size_t)(row0 + t * 16 + l16) * C128 + half * 2;
  // per-lane LDS read column (rotation is constant per lane: odd kp <=> half==1)
  int ncol[8];
#pragma unroll
  for (int nt = 0; nt < 8; ++nt) {
    const int n = nt * 16 + l16;
    ncol[nt] = (half ? ((n + 16) & 127) : n) * 2;
  }
  v2f a[NT];
#pragma unroll
  for (int t = 0; t < NT; ++t) a[t] = *(const v2f*)(arow[t]);
#pragma unroll
  for (int kk = 0; kk < C128; kk += 4) {
    const float* wrow = Wl + (kk / 2 + half) * 256;
    v2f b[8];
#pragma unroll
    for (int nt = 0; nt < 8; ++nt) b[nt] = *(const v2f*)(wrow + ncol[nt]);
    v2f an[NT];
#pragma unroll
    for (int t = 0; t < NT; ++t)
      an[t] = (kk + 4 < C128) ? *(const v2f*)(arow[t] + kk + 4) : a[t];
#pragma unroll
    for (int t = 0; t < NT; ++t)
#pragma unroll
      for (int nt = 0; nt < 8; ++nt)
        acc[t][nt] = __builtin_amdgcn_wmma_f32_16x16x4_f32(
            false, a[t], false, b[nt], (short)0, acc[t][nt], false, false);
#pragma unroll
    for (int t = 0; t < NT; ++t) a[t] = an[t];
  }
}

template <int NT>
__device__ __forceinline__ void store_bias(float* __restrict__ Y,
                                           const float* __restrict__ bias,
                                           const v8f (&acc)[NT][8], int row0, int lane) {
  const int half = lane >> 4;
  const int l16  = lane & 15;
#pragma unroll
  for (int nt = 0; nt < 8; ++nt) {
    const int n = nt * 16 + l16;
    const float bb = bias[n];
#pragma unroll
    for (int t = 0; t < NT; ++t)
#pragma unroll
      for (int j = 0; j < 8; ++j)
        Y[(size_t)(row0 + t * 16 + j + half * 8) * C128 + n] = acc[t][nt][j] + bb;
  }
}

template <int NT>
__device__ __forceinline__ void store_gate_relu(float* __restrict__ Y,
                                                const float* __restrict__ bias,
                                                const float* __restrict__ xres, float g,
                                                const v8f (&acc)[NT][8], int row0, int lane) {
  const int half = lane >> 4;
  const int l16  = lane & 15;
#pragma unroll
  for (int nt = 0; nt < 8; ++nt) {
    const int n = nt * 16 + l16;
    const float bb = bias[n];
#pragma unroll
    for (int t = 0; t < NT; ++t)
#pragma unroll
      for (int j = 0; j < 8; ++j) {
        const size_t r = (size_t)(row0 + t * 16 + j + half * 8);
        const float o  = acc[t][nt][j] + bb;
        const float xr = xres[r * C128 + n];
        const float v  = g * o + (1.0f - g) * xr;
        Y[r * C128 + n] = fmaxf(v, 0.0f);
      }
  }
}

// Y[M x 128] = X[M x 128] @ W[128 x 128] + bias   (M % 16 == 0)
__global__ __launch_bounds__(256)
void gemm_bias_kernel(const float* __restrict__ X, const float* __restrict__ W,
                      const float* __restrict__ bias, float* __restrict__ Y, int M) {
  __shared__ float Wl[WL_FLOATS];
  fill_W_lds(W, Wl, threadIdx.x, blockDim.x);
  __syncthreads();

  const int wave = threadIdx.x >> 5;
  const int lane = threadIdx.x & 31;
  const int row0 = (blockIdx.x * 8 + wave) * 32;
  const v8f zero = {0.f, 0.f, 0.f, 0.f, 0.f, 0.f, 0.f, 0.f};

  if (row0 + 32 <= M) {                    // full: two 16-row tiles, branch-free
    v8f acc[2][8];
#pragma unroll
    for (int t = 0; t < 2; ++t)
#pragma unroll
      for (int i = 0; i < 8; ++i) acc[t][i] = zero;
    wmma_tiles<2>(X, Wl, row0, lane, acc);
    store_bias<2>(Y, bias, acc, row0, lane);
  } else if (row0 < M) {                   // tail: single 16-row tile
    v8f acc[1][8];
#pragma unroll
    for (int i = 0; i < 8; ++i) acc[0][i] = zero;
    wmma_tiles<1>(X, Wl, row0, lane, acc);
    store_bias<1>(Y, bias, acc, row0, lane);
  }
}

// out = relu( g*(X@W + b) + (1-g)*xres ),  g = sigmoid(*skipPtr)
__global__ __launch_bounds__(256)
void gemm_out_kernel(const float* __restrict__ X, const float* __restrict__ W,
                     const float* __restrict__ bias, const float* __restrict__ skipPtr,
                     const float* __restrict__ xres, float* __restrict__ Y, int M) {
  __shared__ float Wl[WL_FLOATS];
  fill_W_lds(W, Wl, threadIdx.x, blockDim.x);
  __syncthreads();

  const int wave = threadIdx.x >> 5;
  const int lane = threadIdx.x & 31;
  const int row0 = (blockIdx.x * 8 + wave) * 32;
  const v8f zero = {0.f, 0.f, 0.f, 0.f, 0.f, 0.f, 0.f, 0.f};
  const float g = 1.0f / (1.0f + expf(-skipPtr[0]));

  if (row0 + 32 <= M) {
    v8f acc[2][8];
#pragma unroll
    for (int t = 0; t < 2; ++t)
#pragma unroll
      for (int i = 0; i < 8; ++i) acc[t][i] = zero;
    wmma_tiles<2>(X, Wl, row0, lane, acc);
    store_gate_relu<2>(Y, bias, xres, g, acc, row0, lane);
  } else if (row0 < M) {
    v8f acc[1][8];
#pragma unroll
    for (int i = 0; i < 8; ++i) acc[0][i] = zero;
    wmma_tiles<1>(X, Wl, row0, lane, acc);
    store_gate_relu<1>(Y, bias, xres, g, acc, row0, lane);
  }
}

// ---------------------------------------------------------------------------
// Fold relation transforms a_rel/m_rel (per-head 64x64) into K/V projection
// weights: Wk'[cin][h*64+e'] = sum_d Wk[cin][h*64+d]*a[h][d][e']; row 128 = bias.
// ---------------------------------------------------------------------------
__global__ void fold_kernel(const float* __restrict__ Wk, const float* __restrict__ bk,
                            const float* __restrict__ Wv, const float* __restrict__ bv,
                            const float* __restrict__ a_rel, const float* __restrict__ m_rel,
                            float* __restrict__ wfold, float* __restrict__ bfold) {
  const int rel_of_type[5] = {3, 4, 0, 1, 2};  // node type -> relation id (as src)
  long id = (long)blockIdx.x * blockDim.x + threadIdx.x;
  const long total = 10L * 129 * 128;
  if (id >= total) return;
  const int col = (int)(id % 128);
  const int row = (int)((id / 128) % 129);
  const int m   = (int)(id / (128L * 129));  // 0..9
  const int t   = m >> 1;
  const bool isV = (m & 1);
  const int e = rel_of_type[t];
  const int h = col >> 6, ep = col & 63;
  const float* Wsrc = (isV ? Wv : Wk) + (size_t)t * 16384;
  const float* bsrc = (isV ? bv : bk) + (size_t)t * 128;
  const float* A = (isV ? m_rel : a_rel) + (size_t)e * HEADS * 64 * 64 + (size_t)h * 64 * 64;
  float s = 0.f;
  if (row < 128) {
    for (int d = 0; d < 64; ++d) s += Wsrc[(size_t)row * 128 + h * 64 + d] * A[d * 64 + ep];
    wfold[(size_t)m * 16384 + (size_t)row * 128 + col] = s;
  } else {
    for (int d = 0; d < 64; ++d) s += bsrc[h * 64 + d] * A[d * 64 + ep];
    bfold[(size_t)m * 128 + col] = s;
  }
}

__global__ void zero_kernel(unsigned* __restrict__ buf, long n) {
  long i = (long)blockIdx.x * blockDim.x + threadIdx.x;
  if (i < n) buf[i] = 0u;
}

// order-preserving float -> uint key (0 acts as -inf sentinel)
__device__ __forceinline__ unsigned fkey(float f) {
  unsigned u = __float_as_uint(f);
  return (u & 0x80000000u) ? ~u : (u | 0x80000000u);
}
__device__ __forceinline__ float fkey_dec(unsigned k) {
  if (k == 0u) return 0.0f;  // empty segment -> reference replaces -inf by 0
  return (k & 0x80000000u) ? __uint_as_float(k & 0x7FFFFFFFu) : __uint_as_float(~k);
}

// ---------------------------------------------------------------------------
// Edge pass 1: one wave per edge.  a[e,h] = (q[di,h,:]·kp[si,h,:]) * p_rel[h]
// * 1/sqrt(64); store logit, atomic-max per (dst,h).
// Lanes 0-15 = head 0 (4 floats each via b128), lanes 16-31 = head 1.
// ---------------------------------------------------------------------------
__global__ void edge_logits_kernel(const int* __restrict__ src, const int* __restrict__ dst,
                                   int E,
                                   const float* __restrict__ q, const float* __restrict__ kp,
                                   const float* __restrict__ prel,
                                   float* __restrict__ logits,       // pre-offset by edge base
                                   unsigned* __restrict__ maxkey, int slotOff) {
  const int wave = (int)(((long)blockIdx.x * blockDim.x + threadIdx.x) >> 5);
  const int lane = threadIdx.x & 31;
  if (wave >= E) return;
  const int half = lane >> 4;
  const int l16  = lane & 15;
  const int si = src[wave];
  const int di = dst[wave];
  const float4 qa = ((const float4*)(q  + (size_t)di * C128 + half * DHEAD))[l16];
  const float4 ka = ((const float4*)(kp + (size_t)si * C128 + half * DHEAD))[l16];
  float p = qa.x * ka.x + qa.y * ka.y + qa.z * ka.z + qa.w * ka.w;
  p += __shfl_xor(p, 1, 32);
  p += __shfl_xor(p, 2, 32);
  p += __shfl_xor(p, 4, 32);
  p += __shfl_xor(p, 8, 32);
  const float a = p * prel[half] * 0.125f;  // 1/sqrt(64)
  if (l16 == 0) {
    logits[(size_t)wave * HEADS + half] = a;
    atomicMax(&maxkey[(size_t)(slotOff + di) * HEADS + half], fkey(a));
  }
}

// ---------------------------------------------------------------------------
// Edge pass 2: e = exp(a - m[dst,h]); agg[dst] += e * vp[src]; sum[dst,h] += e.
// (softmax normalization deferred to the node pass)
// ---------------------------------------------------------------------------
__global__ void edge_scatter_kernel(const int* __restrict__ src, const int* __restrict__ dst,
                                    int E,
                                    const float* __restrict__ vp,
                                    const float* __restrict__ logits,
                                    const unsigned* __restrict__ maxkey,
                                    float* __restrict__ sum, float* __restrict__ agg,
                                    int slotOff) {
  const int wave = (int)(((long)blockIdx.x * blockDim.x + threadIdx.x) >> 5);
  const int lane = threadIdx.x & 31;
  if (wave >= E) return;
  const int half = lane >> 4;
  const int l16  = lane & 15;
  const int si = src[wave];
  const int slot = slotOff + dst[wave];
  const float a = logits[(size_t)wave * HEADS + half];
  const float m = fkey_dec(maxkey[(size_t)slot * HEADS + half]);
  const float ex = expf(a - m);
  const float4 vv = ((const float4*)(vp + (size_t)si * C128 + half * DHEAD))[l16];
  float* ar = agg + (size_t)slot * C128 + half * DHEAD + l16 * 4;
  atomicAdd(ar + 0, ex * vv.x);
  atomicAdd(ar + 1, ex * vv.y);
  atomicAdd(ar + 2, ex * vv.z);
  atomicAdd(ar + 3, ex * vv.w);
  if (l16 == 0) atomicAdd(&sum[(size_t)slot * HEADS + half], ex);
}

// normalize by softmax denominator, exact gelu
__global__ void norm_gelu_kernel(const float* __restrict__ agg, const float* __restrict__ sum,
                                 float* __restrict__ hidden, long total) {
  long i = (long)blockIdx.x * blockDim.x + threadIdx.x;
  if (i >= total) return;
  const long n = i >> 7;
  const int c = (int)(i & 127);
  const int h = c >> 6;
  const float s = fmaxf(sum[n * HEADS + h], 1e-16f);
  const float v = agg[i] / s;
  hidden[i] = 0.5f * v * (1.0f + erff(v * 0.70710678118654752f));
}

// ---------------------------------------------------------------------------
extern "C" void kernel_launch(void* const* d_in, const int* in_sizes, int n_in,
                              void* d_out, int out_size, void* d_ws, size_t ws_size,
                              hipStream_t stream) {
  (void)in_sizes; (void)n_in; (void)out_size; (void)ws_size;
  const int NU = 100000, NI = 50000, NA = 50000;
  const int EA = 50000, EB = 500000;

  const float* x[5]; for (int t = 0; t < 5; ++t) x[t] = (const float*)d_in[t];
  const int* srcp[5] = {(const int*)d_in[5], (const int*)d_in[7], (const int*)d_in[9],
                        (const int*)d_in[11], (const int*)d_in[13]};
  const int* dstp[5] = {(const int*)d_in[6], (const int*)d_in[8], (const int*)d_in[10],
                        (const int*)d_in[12], (const int*)d_in[14]};
  const float* Wq = (const float*)d_in[15]; const float* bq = (const float*)d_in[16];
  const float* Wk = (const float*)d_in[17]; const float* bk = (const float*)d_in[18];
  const float* Wv = (const float*)d_in[19]; const float* bv = (const float*)d_in[20];
  const float* a_rel = (const float*)d_in[21];
  const float* m_rel = (const float*)d_in[22];
  const float* p_rel = (const float*)d_in[23];
  const float* skip  = (const float*)d_in[24];
  const float* Wout  = (const float*)d_in[25];
  const float* bout  = (const float*)d_in[26];
  float* out = (float*)d_out;

  // ---- workspace layout (floats) ----
  float* ws = (float*)d_ws;
  size_t off = 0;
  float* WFOLD = ws + off; off += 10L * 16384;      // folded K'/V' weights
  float* BFOLD = ws + off; off += 10L * 128;        // folded biases
  float* QU = ws + off; off += (size_t)NU * C128;   // q user
  float* QI = ws + off; off += (size_t)NI * C128;   // q item
  float* KP = ws + off; off += 300000L * C128;      // kp all 5 types, concat
  float* VP = ws + off; off += 300000L * C128;      // vp all 5 types, concat
  float* LOGITS = ws + off; off += 1150000L * HEADS;
  unsigned* MAXKEY = (unsigned*)(ws + off); off += 150000L * HEADS;
  float* SUM = ws + off; off += 150000L * HEADS;
  float* AGG = ws + off; off += 150000L * C128;
  float* HIDDEN = ws + off; off += 150000L * C128;

  const size_t kpRow[5] = {0, (size_t)NU, (size_t)NU + NI, (size_t)NU + NI + NA,
                           (size_t)NU + NI + 2 * NA};

  // 1) fold relation transforms into K/V projection weights
  {
    long total = 10L * 129 * 128;
    fold_kernel<<<(unsigned)((total + 255) / 256), 256, 0, stream>>>(
        Wk, bk, Wv, bv, a_rel, m_rel, WFOLD, BFOLD);
  }

  // 2) zero atomic accumulators (MAXKEY, SUM, AGG are contiguous)
  {
    long nz = 150000L * HEADS * 2 + 150000L * C128;
    zero_kernel<<<(unsigned)((nz + 255) / 256), 256, 0, stream>>>(MAXKEY, nz);
  }

  // 3) WMMA projections: q (dst types), kp & vp (all types, folded weights)
  const int Msz[5] = {NU, NI, NA, NA, NA};
#define GEMM_BLOCKS(M) ((unsigned)(((M) + 255) / 256))   // 8 waves x 32 rows per block
  gemm_bias_kernel<<<GEMM_BLOCKS(NU), 256, 0, stream>>>(x[0], Wq + 0, bq + 0, QU, NU);
  gemm_bias_kernel<<<GEMM_BLOCKS(NI), 256, 0, stream>>>(x[1], Wq + 16384, bq + 128, QI, NI);
  for (int t = 0; t < 5; ++t) {
    gemm_bias_kernel<<<GEMM_BLOCKS(Msz[t]), 256, 0, stream>>>(
        x[t], WFOLD + (size_t)(2 * t) * 16384, BFOLD + (size_t)(2 * t) * 128,
        KP + kpRow[t] * C128, Msz[t]);
    gemm_bias_kernel<<<GEMM_BLOCKS(Msz[t]), 256, 0, stream>>>(
        x[t], WFOLD + (size_t)(2 * t + 1) * 16384, BFOLD + (size_t)(2 * t + 1) * 128,
        VP + kpRow[t] * C128, Msz[t]);
  }

  // 4) edge metadata: (srcType, E, q, slotOff, edgeBase) per relation
  const int eSrcT[5] = {2, 3, 4, 0, 1};
  const int eCnt[5]  = {EA, EA, EA, EB, EB};
  const int eSlot[5] = {NU, NU, NU, NU, 0};  // dst item -> slots [NU,150k); dst user -> [0,NU)
  const size_t eBase[5] = {0, (size_t)EA, 2 * (size_t)EA, 3 * (size_t)EA,
                           3 * (size_t)EA + EB};
  // pass 1: logits + segment max
  for (int e = 0; e < 5; ++e) {
    const float* qarr = (e == 4) ? QU : QI;
    unsigned blocks = (unsigned)((eCnt[e] + 7) / 8);  // 8 waves/block, 1 edge/wave
    edge_logits_kernel<<<blocks, 256, 0, stream>>>(
        srcp[e], dstp[e], eCnt[e], qarr, KP + kpRow[eSrcT[e]] * C128,
        p_rel + (size_t)e * HEADS, LOGITS + eBase[e] * HEADS, MAXKEY, eSlot[e]);
  }
  // pass 2: exp + weighted scatter + denominator
  for (int e = 0; e < 5; ++e) {
    unsigned blocks = (unsigned)((eCnt[e] + 7) / 8);
    edge_scatter_kernel<<<blocks, 256, 0, stream>>>(
        srcp[e], dstp[e], eCnt[e], VP + kpRow[eSrcT[e]] * C128,
        LOGITS + eBase[e] * HEADS, MAXKEY, SUM, AGG, eSlot[e]);
  }

  // 5) normalize + gelu
  {
    long total = 150000L * C128;
    norm_gelu_kernel<<<(unsigned)((total + 255) / 256), 256, 0, stream>>>(
        AGG, SUM, HIDDEN, total);
  }

  // 6) output projection + skip gate + relu (WMMA), users then items into d_out
  gemm_out_kernel<<<GEMM_BLOCKS(NU), 256, 0, stream>>>(
      HIDDEN, Wout + 0, bout + 0, skip + 0, x[0], out, NU);
  gemm_out_kernel<<<GEMM_BLOCKS(NI), 256, 0, stream>>>(
      HIDDEN + (size_t)NU * C128, Wout + 16384, bout + 128, skip + 1, x[1],
      out + (size_t)NU * C128, NI);
#undef GEMM_BLOCKS
}